// TpconvLayer_71992241815600
// MI455X (gfx1250) — compile-verified
//
#include <hip/hip_runtime.h>

typedef __attribute__((ext_vector_type(2))) float v2f;
typedef __attribute__((ext_vector_type(8))) float v8f;
typedef __attribute__((ext_vector_type(4))) unsigned int v4u;
typedef __attribute__((ext_vector_type(4))) int v4i;
typedef __attribute__((ext_vector_type(8))) int v8i;

#define MUL0 64
#define MUL1 32
#define DIM  160
#define NNODES 50000
#define NEDGES 800000

// N0 = 1/sqrt(2*MUL0) = 1/sqrt(128), N1 = 1/sqrt(2*MUL1) = 1/8
#define N0C 0.08838834764831845f
#define N1C 0.125f

#if __has_include(<hip/amd_detail/amd_gfx1250_TDM.h>)
#define TDM_6ARG 1
#endif

// ---- per-wave LDS layout (floats) ----
// sT   : 16 rows x 165 dwords (160 data + 5 TDM pad dwords; pad every 32 dw)
//        element (e, j) at e*165 + j + (j>>5)
// sDot : 16 rows x 36 (32 used, padded)
// sU   : 16 x 4
// sDst : 16 ints ; sSrc : 16 ints (fallback path only)
#define ROWP     165
#define TILE_F   (16 * ROWP)        // 2640
#define DOT_OFF  TILE_F             // 2640
#define U_OFF    (DOT_OFF + 16*36)  // 3216
#define DST_OFF  (U_OFF + 64)       // 3280
#define SRC_OFF  (DST_OFF + 16)     // 3296
#define WAVE_FLOATS 3312
#define WAVES_PER_BLOCK 4

__global__ void residual_copy_kernel(const float* __restrict__ atom,
                                     float* __restrict__ out, int n4) {
    int i = blockIdx.x * blockDim.x + threadIdx.x;
    if (i < n4) {
        ((float4*)out)[i] = ((const float4*)atom)[i];
    }
}

__global__ __launch_bounds__(WAVES_PER_BLOCK * 32)
void tpconv_edge_kernel(const float* __restrict__ atom,
                        const float* __restrict__ ev,
                        const int*   __restrict__ ei,
                        const float* __restrict__ w00,
                        const float* __restrict__ w11,
                        const float* __restrict__ w01,
                        const float* __restrict__ w10,
                        float* __restrict__ out) {
    __shared__ float smem[WAVES_PER_BLOCK * WAVE_FLOATS];

    const int lane = threadIdx.x & 31;
    const int wave = threadIdx.x >> 5;
    float* Wm   = smem + wave * WAVE_FLOATS;
    float* sT   = Wm;             // raw gathered rows (padded)
    float* sDot = Wm + DOT_OFF;
    float* sU   = Wm + U_OFF;
    int*   sDst = (int*)(Wm + DST_OFF);
    int*   sSrc = (int*)(Wm + SRC_OFF);

    const int eBase = (blockIdx.x * WAVES_PER_BLOCK + wave) * 16;

    // ---- 1) edge meta: normalize u, fetch dst (and src for fallback) ----
    if (lane < 16) {
        int e = eBase + lane;
        float ex = ev[e * 3 + 0];
        float ey = ev[e * 3 + 1];
        float ez = ev[e * 3 + 2];
        float inv = 1.0f / (sqrtf(ex * ex + ey * ey + ez * ez) + 1e-9f);
        sU[lane * 4 + 0] = ex * inv;
        sU[lane * 4 + 1] = ey * inv;
        sU[lane * 4 + 2] = ez * inv;
        sDst[lane] = ei[NEDGES + e];
        sSrc[lane] = ei[e];
    }

#if __has_builtin(__builtin_amdgcn_tensor_load_to_lds)
    // ---- 2) TDM gather-mode: fetch 16 src rows (160 f32 each) into LDS ----
    {
        int eb = __builtin_amdgcn_readfirstlane(eBase);
        const int* sp = ei + eb;
        unsigned int pk[8];
#pragma unroll
        for (int j = 0; j < 8; ++j) {
            unsigned int i0 = (unsigned int)__builtin_amdgcn_readfirstlane(sp[2 * j]);
            unsigned int i1 = (unsigned int)__builtin_amdgcn_readfirstlane(sp[2 * j + 1]);
            pk[j] = (i0 & 0xFFFFu) | ((i1 & 0xFFFFu) << 16);
        }
        unsigned int lds_b =
            (unsigned int)__builtin_amdgcn_readfirstlane((int)(uintptr_t)(void*)sT);
        unsigned long long ga = (unsigned long long)(uintptr_t)atom;

        // Group 0: gather_mode=1, 16-bit indices, count=1, lds_addr, global_addr, type=2
        v4u g0;
        g0.x = 0x80000001u;
        g0.y = lds_b;
        g0.z = (unsigned int)(ga & 0xFFFFFFFFu);
        g0.w = (unsigned int)((ga >> 32) & 0x01FFFFFFu) | 0x80000000u;
        // Group 1: data_size=4B, pad every 32 dwords by 1 dword,
        // tensor_dim0=160, tensor_dim1=50000, tile_dim0=160, tile_dim1=16 idx,
        // tensor_dim0_stride=160
        v8i g1;
        g1[0] = 0x01120000;               // data_size=2 | pad_en | pad_interval=4
        g1[1] = (int)(160u << 16);        // tensor_dim0 low16 in [31:16]
        g1[2] = (int)0xC3500000u;         // tensor_dim1 low16 (50000) in [31:16]
        g1[3] = (int)(160u << 16);        // tile_dim0 in [31:16]
        g1[4] = 16;                       // tile_dim1 = #valid indices
        g1[5] = 160;                      // tensor_dim0_stride low32
        g1[6] = 0;
        g1[7] = 0;
        v4i g2, g3;
        g2[0] = (int)pk[0]; g2[1] = (int)pk[1]; g2[2] = (int)pk[2]; g2[3] = (int)pk[3];
        g3[0] = (int)pk[4]; g3[1] = (int)pk[5]; g3[2] = (int)pk[6]; g3[3] = (int)pk[7];
#if defined(TDM_6ARG)
        v8i gx = {0, 0, 0, 0, 0, 0, 0, 0};
        __builtin_amdgcn_tensor_load_to_lds(g0, g1, g2, g3, gx, 0);
#else
        __builtin_amdgcn_tensor_load_to_lds(g0, g1, g2, g3, 0);
#endif
    }
    __syncthreads();
    __builtin_amdgcn_s_wait_tensorcnt(0);
    asm volatile("" ::: "memory");
#else
    // ---- 2) fallback: manual gather into the same padded layout ----
    __syncthreads();
#pragma unroll
    for (int i = 0; i < 20; ++i) {
        int v4  = i * 32 + lane;          // 0..639 float4s
        int row = v4 / 40;                // 40 float4 per 160-f32 row
        int c4  = v4 % 40;
        int src = sSrc[row];
        float4 d = *(const float4*)(atom + (size_t)src * DIM + c4 * 4);
        float vals[4] = {d.x, d.y, d.z, d.w};
#pragma unroll
        for (int t = 0; t < 4; ++t) {
            int j = c4 * 4 + t;
            sT[row * ROWP + j + (j >> 5)] = vals[t];
        }
    }
    __syncthreads();
#endif

    // ---- 3) dot[e][m] = (v[e][m] . u[e]) * N1  ----
    {
        int e  = lane & 15;
        int mb = (lane >> 4) * 16;
        float u0 = sU[e * 4 + 0], u1 = sU[e * 4 + 1], u2 = sU[e * 4 + 2];
        const float* row = sT + e * ROWP;
#pragma unroll
        for (int m = mb; m < mb + 16; ++m) {
            int j0 = 64 + 3 * m, j1 = j0 + 1, j2 = j0 + 2;
            float dd = row[j0 + (j0 >> 5)] * u0
                     + row[j1 + (j1 >> 5)] * u1
                     + row[j2 + (j2 >> 5)] * u2;
            sDot[e * 36 + m] = dd * N1C;
        }
    }
    __syncthreads();

    // A fragment addressing (16x4 f32 A): lane<16 -> M=lane,K=4kk+{0,1};
    //                                     lane>=16 -> M=lane-16,K=4kk+{2,3}
    const int mA   = lane & 15;
    const int kOff = (lane >> 4) * 2;
    const int nB   = lane & 15;
    const int nLoc = lane & 15;
    const int mAdd = (lane >> 4) * 8;
    const float* aRow = sT + mA * ROWP;

    // ---- Phase A: m0 = S @ (w00*N0) + dotN1 @ w11 ----
    v8f acc0[4] = {v8f{0}, v8f{0}, v8f{0}, v8f{0}};
#pragma unroll 4
    for (int kk = 0; kk < 16; ++kk) {
        int k0 = 4 * kk + kOff;
        v2f a;
        a.x = aRow[k0 + (k0 >> 5)];
        a.y = aRow[k0 + 1 + ((k0 + 1) >> 5)];
#pragma unroll
        for (int nt = 0; nt < 4; ++nt) {
            v2f b;
            b.x = w00[(k0) * MUL0 + nt * 16 + nB] * N0C;
            b.y = w00[(k0 + 1) * MUL0 + nt * 16 + nB] * N0C;
            acc0[nt] = __builtin_amdgcn_wmma_f32_16x16x4_f32(
                false, a, false, b, (short)0, acc0[nt], false, false);
        }
    }
#pragma unroll 4
    for (int kk = 0; kk < 8; ++kk) {
        int k0 = 4 * kk + kOff;
        v2f a;
        a.x = sDot[mA * 36 + k0];
        a.y = sDot[mA * 36 + k0 + 1];
#pragma unroll
        for (int nt = 0; nt < 4; ++nt) {
            v2f b;
            b.x = w11[(k0) * MUL0 + nt * 16 + nB];
            b.y = w11[(k0 + 1) * MUL0 + nt * 16 + nB];
            acc0[nt] = __builtin_amdgcn_wmma_f32_16x16x4_f32(
                false, a, false, b, (short)0, acc0[nt], false, false);
        }
    }
    // scatter m0
#pragma unroll
    for (int nt = 0; nt < 4; ++nt) {
#pragma unroll
        for (int r = 0; r < 8; ++r) {
            int e   = r + mAdd;
            int dst = sDst[e];
            atomicAdd(out + (size_t)dst * DIM + nt * 16 + nLoc, acc0[nt][r]);
        }
    }

    // ---- Phase B: p = S @ (w01*N0) ----
    v8f acc1[2] = {v8f{0}, v8f{0}};
#pragma unroll 4
    for (int kk = 0; kk < 16; ++kk) {
        int k0 = 4 * kk + kOff;
        v2f a;
        a.x = aRow[k0 + (k0 >> 5)];
        a.y = aRow[k0 + 1 + ((k0 + 1) >> 5)];
#pragma unroll
        for (int nt = 0; nt < 2; ++nt) {
            v2f b;
            b.x = w01[(k0) * MUL1 + nt * 16 + nB] * N0C;
            b.y = w01[(k0 + 1) * MUL1 + nt * 16 + nB] * N0C;
            acc1[nt] = __builtin_amdgcn_wmma_f32_16x16x4_f32(
                false, a, false, b, (short)0, acc1[nt], false, false);
        }
    }

    // ---- Phase C: per c: q = V_c @ (w10*N1) ; m1 = q + p*u_c ----
#pragma unroll
    for (int c = 0; c < 3; ++c) {
        v8f q[2] = {v8f{0}, v8f{0}};
#pragma unroll 4
        for (int kk = 0; kk < 8; ++kk) {
            int k0 = 4 * kk + kOff;           // K index = m
            int j0 = 64 + 3 * k0 + c;
            int j1 = j0 + 3;
            v2f a;
            a.x = aRow[j0 + (j0 >> 5)];
            a.y = aRow[j1 + (j1 >> 5)];
#pragma unroll
            for (int nt = 0; nt < 2; ++nt) {
                v2f b;
                b.x = w10[(k0) * MUL1 + nt * 16 + nB] * N1C;
                b.y = w10[(k0 + 1) * MUL1 + nt * 16 + nB] * N1C;
                q[nt] = __builtin_amdgcn_wmma_f32_16x16x4_f32(
                    false, a, false, b, (short)0, q[nt], false, false);
            }
        }
#pragma unroll
        for (int nt = 0; nt < 2; ++nt) {
#pragma unroll
            for (int r = 0; r < 8; ++r) {
                int e   = r + mAdd;
                int dst = sDst[e];
                float uc = sU[e * 4 + c];
                float val = q[nt][r] + acc1[nt][r] * uc;
                atomicAdd(out + (size_t)dst * DIM + MUL0 + (nt * 16 + nLoc) * 3 + c, val);
            }
        }
    }
}

extern "C" void kernel_launch(void* const* d_in, const int* in_sizes, int n_in,
                              void* d_out, int out_size, void* d_ws, size_t ws_size,
                              hipStream_t stream) {
    const float* atom = (const float*)d_in[0];
    const float* ev   = (const float*)d_in[1];
    const int*   ei   = (const int*)d_in[2];
    const float* w00  = (const float*)d_in[3];
    const float* w11  = (const float*)d_in[4];
    const float* w01  = (const float*)d_in[5];
    const float* w10  = (const float*)d_in[6];
    float* out = (float*)d_out;

    // 1) out = atom_feature (residual), vectorized copy
    int n4 = (NNODES * DIM) / 4;  // 2,000,000 float4
    residual_copy_kernel<<<(n4 + 255) / 256, 256, 0, stream>>>(atom, out, n4);

    // 2) edge messages + scatter-add. 800000 / (4 waves * 16) = 12500 blocks exact.
    int blocks = NEDGES / (WAVES_PER_BLOCK * 16);
    tpconv_edge_kernel<<<blocks, WAVES_PER_BLOCK * 32, 0, stream>>>(
        atom, ev, ei, w00, w11, w01, w10, out);
}